// ProbAttention_17721035063910
// MI455X (gfx1250) — compile-verified
//
#include <hip/hip_runtime.h>
#include <hip/hip_bf16.h>
#include <math.h>

// Problem constants (from reference setup_inputs)
#define Bn 4
#define Ln 4096
#define Hn 8
#define Dn 64
#define SK 9     // sample_k = ceil(log(4096))
#define Un 9     // u = top-k count
#define BH (Bn*Hn)
#define VCH 32   // L-chunks for V-mean partials (128 rows each)
#define KCH 8    // K-chunks for ctx WMMA partials (512 k each)

typedef __attribute__((ext_vector_type(2))) float v2f;
typedef __attribute__((ext_vector_type(4))) float v4f;
typedef __attribute__((ext_vector_type(8))) float v8f;

// ---------------------------------------------------------------------------
// Kernel 1: M[bh*L + l] = max_s(QK_s) - sum_s(QK_s)/L  (one wave per query)
// ---------------------------------------------------------------------------
__global__ void pa_sparsity_m(const float* __restrict__ Q,
                              const float* __restrict__ K,
                              const int*   __restrict__ idxs,
                              float* __restrict__ M) {
    int wid  = (blockIdx.x * blockDim.x + threadIdx.x) >> 5;
    int lane = threadIdx.x & 31;
    if (wid >= BH * Ln) return;
    int bh = wid >> 12;          // /4096
    int l  = wid & (Ln - 1);
    int b  = bh >> 3, h = bh & 7;
    const float* q = Q + (((b * Ln + l) * Hn + h) * Dn);
    float q0 = q[lane], q1 = q[lane + 32];
    float mx = -3.4e38f, sm = 0.f;
    for (int s = 0; s < SK; ++s) {
        int ks = idxs[l * SK + s];
        const float* kr = K + (((b * Ln + ks) * Hn + h) * Dn);
        float p = q0 * kr[lane] + q1 * kr[lane + 32];
        for (int off = 16; off > 0; off >>= 1) p += __shfl_xor(p, off);
        mx = fmaxf(mx, p);
        sm += p;
    }
    if (lane == 0) M[wid] = mx - sm * (1.0f / (float)Ln);
}

// ---------------------------------------------------------------------------
// Kernel 2: top-9 indices of M per (b,h), descending, lower-index tie-break.
// ---------------------------------------------------------------------------
__global__ void pa_topk(const float* __restrict__ M, int* __restrict__ Mtop) {
    __shared__ float sv[Ln];
    __shared__ float rv[256];
    __shared__ int   ri[256];
    int bh = blockIdx.x, tid = threadIdx.x;
    for (int i = tid; i < Ln; i += 256) sv[i] = M[bh * Ln + i];
    __syncthreads();
    for (int j = 0; j < Un; ++j) {
        float bv = -3.4e38f; int bi = 0x7fffffff;
        for (int i = tid; i < Ln; i += 256) {
            float v = sv[i];
            if (v > bv) { bv = v; bi = i; }
        }
        rv[tid] = bv; ri[tid] = bi;
        __syncthreads();
        for (int s = 128; s > 0; s >>= 1) {
            if (tid < s) {
                float ov = rv[tid + s]; int oi = ri[tid + s];
                if (ov > rv[tid] || (ov == rv[tid] && oi < ri[tid])) { rv[tid] = ov; ri[tid] = oi; }
            }
            __syncthreads();
        }
        if (tid == 0) { Mtop[bh * 16 + j] = ri[0]; sv[ri[0]] = -3.4e38f; }
        __syncthreads();
    }
    if (tid < 16 - Un) Mtop[bh * 16 + Un + tid] = 0;  // pad rows 9..15
}

// ---------------------------------------------------------------------------
// Kernel 3: gather Q_reduce (bh,16,64), rows >= 9 zeroed.
// ---------------------------------------------------------------------------
__global__ void pa_gather_q(const float* __restrict__ Q,
                            const int*   __restrict__ Mtop,
                            float* __restrict__ Qred) {
    int bh = blockIdx.x, d = threadIdx.x;
    int b = bh >> 3, h = bh & 7;
    for (int m = 0; m < 16; ++m) {
        float v = 0.f;
        if (m < Un) {
            int pos = Mtop[bh * 16 + m];
            v = Q[(((b * Ln + pos) * Hn + h) * Dn) + d];
        }
        Qred[(bh * 16 + m) * Dn + d] = v;
    }
}

// ---------------------------------------------------------------------------
// Kernel 4a: partial V sums over 128-row chunks of L (bh x 32 chunks).
// ---------------------------------------------------------------------------
__global__ void pa_vmean_part(const float* __restrict__ V, float* __restrict__ Vpart) {
    int blk = blockIdx.x;            // bh*VCH + chunk
    int bh = blk / VCH, ck = blk % VCH;
    int d = threadIdx.x;
    int b = bh >> 3, h = bh & 7;
    int l0 = ck * (Ln / VCH);
    float s = 0.f;
    for (int l = l0; l < l0 + Ln / VCH; ++l)
        s += V[(((b * Ln + l) * Hn + h) * Dn) + d];
    Vpart[(bh * VCH + ck) * Dn + d] = s;
}

// Kernel 4b: reduce partials -> Vmean
__global__ void pa_vmean_reduce(const float* __restrict__ Vpart, float* __restrict__ Vm) {
    int bh = blockIdx.x, d = threadIdx.x;
    float s = 0.f;
    for (int ck = 0; ck < VCH; ++ck) s += Vpart[(bh * VCH + ck) * Dn + d];
    Vm[bh * Dn + d] = s * (1.0f / (float)Ln);
}

// ---------------------------------------------------------------------------
// Kernel 5: broadcast Vmean into context output (B,L,H,D), float4 stores.
// ---------------------------------------------------------------------------
__global__ void pa_bcast_ctx(const float* __restrict__ Vm, float* __restrict__ out) {
    const int n4 = (Bn * Ln * Hn * Dn) / 4;
    const v4f* vm4 = (const v4f*)Vm;
    v4f* out4 = (v4f*)out;
    for (int i = blockIdx.x * blockDim.x + threadIdx.x; i < n4; i += gridDim.x * blockDim.x) {
        int d4 = i & 15;             // 16 float4s per D=64
        int h  = (i >> 4) & 7;
        int b  = i >> 19;
        out4[i] = vm4[(((b << 3) + h) << 4) + d4];
    }
}

// ---------------------------------------------------------------------------
// Kernel 6: scores = Qred(16x64) . K^T(64x4096) * 0.125 via v_wmma_f32_16x16x4_f32
// One wave per (bh, 16-column tile). Rows 0..8 written to attn region.
// ---------------------------------------------------------------------------
__global__ void pa_scores_wmma(const float* __restrict__ Qred,
                               const float* __restrict__ K,
                               float* __restrict__ attn) {
    int wid  = (blockIdx.x * blockDim.x + threadIdx.x) >> 5;
    int lane = threadIdx.x & 31;
    int bh = wid >> 8;          // /256 tiles
    int nt = wid & 255;
    int b = bh >> 3, h = bh & 7;
    int mn = lane & 15;         // row (A) / col (B,C)
    int hi = lane >> 4;         // half-wave select
    v8f c = {};
    for (int kk = 0; kk < 16; ++kk) {          // K = 64 in steps of 4
        int kb = kk * 4 + hi * 2;
        v2f a, bb;
        const float* qr = Qred + (bh * 16 + mn) * Dn + kb;
        a.x = qr[0]; a.y = qr[1];
        const float* kr = K + (((b * Ln + nt * 16 + mn) * Hn + h) * Dn) + kb;
        bb.x = kr[0]; bb.y = kr[1];
        c = __builtin_amdgcn_wmma_f32_16x16x4_f32(false, a, false, bb,
                                                  (short)0, c, false, false);
    }
#pragma unroll
    for (int v = 0; v < 8; ++v) {
        int row = v + 8 * hi;
        if (row < Un)
            attn[(bh * Un + row) * Ln + nt * 16 + mn] = c[v] * 0.125f;
    }
}

// ---------------------------------------------------------------------------
// Kernel 7: in-place softmax over each attn row (length 4096).
// ---------------------------------------------------------------------------
__global__ void pa_softmax(float* __restrict__ attn) {
    __shared__ float red[256];
    int row = blockIdx.x; int tid = threadIdx.x;
    float* rp = attn + row * Ln;
    float mx = -3.4e38f;
    for (int i = tid; i < Ln; i += 256) mx = fmaxf(mx, rp[i]);
    red[tid] = mx; __syncthreads();
    for (int s = 128; s > 0; s >>= 1) { if (tid < s) red[tid] = fmaxf(red[tid], red[tid + s]); __syncthreads(); }
    mx = red[0]; __syncthreads();
    float sm = 0.f;
    for (int i = tid; i < Ln; i += 256) { float e = expf(rp[i] - mx); rp[i] = e; sm += e; }
    red[tid] = sm; __syncthreads();
    for (int s = 128; s > 0; s >>= 1) { if (tid < s) red[tid] += red[tid + s]; __syncthreads(); }
    float inv = 1.0f / red[0]; __syncthreads();
    for (int i = tid; i < Ln; i += 256) rp[i] *= inv;
}

// ---------------------------------------------------------------------------
// Kernel 8a: partial ctx_top = attn(16x512) . V(512x64) per K-chunk via WMMA.
// One wave per (bh, d-tile, k-chunk) -> 32*4*8 = 1024 waves. Full 16x16 tile
// written to Cpart.
// ---------------------------------------------------------------------------
__global__ void pa_ctx_wmma(const float* __restrict__ attn,
                            const float* __restrict__ V,
                            float* __restrict__ Cpart) {
    int wid  = (blockIdx.x * blockDim.x + threadIdx.x) >> 5;
    int lane = threadIdx.x & 31;
    int ck = wid & (KCH - 1);
    int dt = (wid >> 3) & 3;
    int bh = wid >> 5;
    int b = bh >> 3, h = bh & 7;
    int mn = lane & 15;
    int hi = lane >> 4;
    const int kper = Ln / KCH;   // 512
    v8f c = {};
    for (int kb = ck * kper; kb < (ck + 1) * kper; kb += 4) {
        int k0 = kb + hi * 2;
        v2f a, bb;
        if (mn < Un) {
            const float* ap = attn + (bh * Un + mn) * Ln + k0;
            a.x = ap[0]; a.y = ap[1];
        } else { a.x = 0.f; a.y = 0.f; }
        bb.x = V[(((b * Ln + k0)     * Hn + h) * Dn) + dt * 16 + mn];
        bb.y = V[(((b * Ln + k0 + 1) * Hn + h) * Dn) + dt * 16 + mn];
        c = __builtin_amdgcn_wmma_f32_16x16x4_f32(false, a, false, bb,
                                                  (short)0, c, false, false);
    }
    float* tile = Cpart + (size_t)(((bh * 4 + dt) * KCH) + ck) * 256;
#pragma unroll
    for (int v = 0; v < 8; ++v) {
        int row = v + 8 * hi;
        tile[row * 16 + mn] = c[v];
    }
}

// Kernel 8b: reduce K-chunk partials and scatter rows 0..8 into context.
// grid: 32 bh * 4 d-tiles = 128 blocks, 144 active threads (9 rows x 16 cols).
__global__ void pa_ctx_reduce(const float* __restrict__ Cpart,
                              const int*   __restrict__ Mtop,
                              float* __restrict__ out) {
    int blk = blockIdx.x;          // bh*4 + dt
    int bh = blk >> 2, dt = blk & 3;
    int b = bh >> 3, h = bh & 7;
    int tid = threadIdx.x;
    if (tid >= Un * 16) return;
    int m = tid >> 4, n = tid & 15;
    const float* base = Cpart + (size_t)(blk * KCH) * 256 + m * 16 + n;
    float s = 0.f;
#pragma unroll
    for (int ck = 0; ck < KCH; ++ck) s += base[ck * 256];
    int pos = Mtop[bh * 16 + m];
    out[(((b * Ln + pos) * Hn + h) * Dn) + dt * 16 + n] = s;
}

// ---------------------------------------------------------------------------
extern "C" void kernel_launch(void* const* d_in, const int* in_sizes, int n_in,
                              void* d_out, int out_size, void* d_ws, size_t ws_size,
                              hipStream_t stream) {
    const float* Q    = (const float*)d_in[0];  // (B,L,H,D)
    const float* K    = (const float*)d_in[1];
    const float* V    = (const float*)d_in[2];
    const int*   idxs = (const int*)  d_in[3];  // (L, 9)

    float* out  = (float*)d_out;
    float* attn = out + (size_t)Bn * Ln * Hn * Dn;   // attn region of d_out

    // Workspace layout (floats): M | Vmean | Vpart | Qred | Cpart | Mtop(int)
    float* Mbuf  = (float*)d_ws;                     // BH*L            = 131072
    float* Vm    = Mbuf  + BH * Ln;                  // BH*D            = 2048
    float* Vpart = Vm    + BH * Dn;                  // BH*VCH*D        = 65536
    float* Qred  = Vpart + BH * VCH * Dn;            // BH*16*D         = 32768
    float* Cpart = Qred  + BH * 16 * Dn;             // BH*4*KCH*256    = 262144
    int*   Mtop  = (int*)(Cpart + BH * 4 * KCH * 256);

    pa_sparsity_m  <<<(BH * Ln) / 8, 256, 0, stream>>>(Q, K, idxs, Mbuf);
    pa_topk        <<<BH, 256, 0, stream>>>(Mbuf, Mtop);
    pa_gather_q    <<<BH, 64, 0, stream>>>(Q, Mtop, Qred);
    pa_vmean_part  <<<BH * VCH, 64, 0, stream>>>(V, Vpart);
    pa_vmean_reduce<<<BH, 64, 0, stream>>>(Vpart, Vm);
    pa_bcast_ctx   <<<2048, 256, 0, stream>>>(Vm, out);
    // 32 bh * 256 tiles = 8192 waves, 8 waves/block
    pa_scores_wmma <<<1024, 256, 0, stream>>>(Qred, K, attn);
    pa_softmax     <<<BH * Un, 256, 0, stream>>>(attn);
    // 32 bh * 4 d-tiles * 8 k-chunks = 1024 waves, 8 waves/block
    pa_ctx_wmma    <<<128, 256, 0, stream>>>(attn, V, Cpart);
    pa_ctx_reduce  <<<BH * 4, 160, 0, stream>>>(Cpart, Mtop, out);
}